// LIFSpike_6021544149124
// MI455X (gfx1250) — compile-verified
//
#include <hip/hip_runtime.h>
#include <cstdint>

// LIF spiking neuron forward (T=8, leak=1.0, v_th=1.0, hard reset).
// Memory-bound streaming kernel: 128 MiB in + 128 MiB out, ~11.5 us at 23.3 TB/s.
// Data path: global_load_async_to_lds_b128 (ASYNCcnt, gfx1250) -> LDS -> recurrence
// in VGPRs -> nontemporal global_store_b128.

#define T_STEPS 8

typedef float v4f __attribute__((ext_vector_type(4)));

__global__ __launch_bounds__(256) void lif_spike_async_kernel(
    const float* __restrict__ x, float* __restrict__ out, int nvec /* float4 per timestep */) {

    // Dynamic LDS only (no static LDS) => dynamic LDS base offset is 0, so raw
    // byte offsets used in the async-load asm match the compiler's addressing.
    extern __shared__ v4f smem[];  // layout: smem[tid*8 + t], 32 KiB per block

    const int tid = threadIdx.x;
    const int idx = blockIdx.x * blockDim.x + tid;
    const bool active = idx < nvec;

    const size_t step = (size_t)nvec;                       // float4 stride between timesteps
    const uint32_t lds_base = (uint32_t)tid * (T_STEPS * 16u);

    // Issue all 8 independent timestep loads as async global->LDS B128 copies.
    // Each lane writes only its own LDS slots, so the per-wave ASYNCcnt wait
    // below is sufficient (no workgroup barrier needed).
    if (active) {
        const v4f* gp = (const v4f*)x + idx;
#pragma unroll
        for (int t = 0; t < T_STEPS; ++t) {
            uint64_t ga = (uint64_t)(uintptr_t)(gp + (size_t)t * step);
            uint32_t la = lds_base + (uint32_t)t * 16u;
            asm volatile("global_load_async_to_lds_b128 %0, %1, off"
                         :: "v"(la), "v"(ga)
                         : "memory");
        }
    }
    asm volatile("s_wait_asynccnt 0x0" ::: "memory");

    if (active) {
        float vx = 0.0f, vy = 0.0f, vz = 0.0f, vw = 0.0f;
        v4f* op = (v4f*)out + idx;
#pragma unroll
        for (int t = 0; t < T_STEPS; ++t) {
            v4f xt = smem[tid * T_STEPS + t];  // ds_load_b128

            vx += xt.x; vy += xt.y; vz += xt.z; vw += xt.w;

            v4f s;
            // s = (v - 1 >= 0); hard reset v = (1 - s) * v
            s.x = (vx >= 1.0f) ? 1.0f : 0.0f;  vx = (vx >= 1.0f) ? 0.0f : vx;
            s.y = (vy >= 1.0f) ? 1.0f : 0.0f;  vy = (vy >= 1.0f) ? 0.0f : vy;
            s.z = (vz >= 1.0f) ? 1.0f : 0.0f;  vz = (vz >= 1.0f) ? 0.0f : vz;
            s.w = (vw >= 1.0f) ? 1.0f : 0.0f;  vw = (vw >= 1.0f) ? 0.0f : vw;

            // Write-once output, never re-read: non-temporal B128 store.
            __builtin_nontemporal_store(s, op + (size_t)t * step);
        }
    }
}

extern "C" void kernel_launch(void* const* d_in, const int* in_sizes, int n_in,
                              void* d_out, int out_size, void* d_ws, size_t ws_size,
                              hipStream_t stream) {
    const float* x = (const float*)d_in[0];
    float* out = (float*)d_out;

    const int n = in_sizes[0];            // 256*128*32*32 = 33,554,432 floats
    const int per_t = n / T_STEPS;        // elements per timestep (4,194,304)
    const int nvec = per_t / 4;           // float4 elements per timestep (1,048,576)

    const int block = 256;
    const int grid = (nvec + block - 1) / block;          // 4096 blocks
    const size_t shmem = (size_t)block * T_STEPS * 16u;   // 32 KiB

    lif_spike_async_kernel<<<grid, block, shmem, stream>>>(x, out, nvec);
}